// myGCN1_27032524161265
// MI455X (gfx1250) — compile-verified
//
#include <hip/hip_runtime.h>
#include <math.h>

typedef __attribute__((ext_vector_type(2))) float v2f;
typedef __attribute__((ext_vector_type(8))) float v8f;

#define GCN_N   100000
#define GCN_E   1600000
#define NFEAT   128
#define NHID    64
#define NCLASS  40
#define MTILES  (GCN_N / 16)          // 6250 row tiles, exact
#define WPB     5                     // waves per block
#define TPB     (WPB * 32)            // 160 threads; 6250/5 = 1250 blocks exact

__device__ __forceinline__ v8f wmma_f32_4(v2f a, v2f b, v8f c) {
    // 8 args: (neg_a, A, neg_b, B, c_mod, C, reuse_a, reuse_b)
    return __builtin_amdgcn_wmma_f32_16x16x4_f32(false, a, false, b, (short)0, c,
                                                 false, false);
}

// ---- degree / normalization -------------------------------------------------
__global__ void gcn_deg_init(float* __restrict__ deg) {
    int i = blockIdx.x * blockDim.x + threadIdx.x;
    if (i < GCN_N) deg[i] = 1.0f;                       // self loop
}

__global__ void gcn_deg_edges(const int* __restrict__ dst, float* __restrict__ deg) {
    int e = blockIdx.x * blockDim.x + threadIdx.x;
    if (e < GCN_E) atomicAdd(&deg[dst[e]], 1.0f);
}

__global__ void gcn_dinv(float* __restrict__ deg) {
    int i = blockIdx.x * blockDim.x + threadIdx.x;
    if (i < GCN_N) deg[i] = rsqrtf(deg[i]);             // deg >= 1 always
}

// ---- GEMM1: h = x @ W1  (fp32 WMMA, one wave per 16-row tile) ---------------
__global__ void __launch_bounds__(TPB)
gcn_gemm1(const float* __restrict__ x, const float* __restrict__ W1,
          float* __restrict__ h) {
    const int wave = blockIdx.x * WPB + (threadIdx.x >> 5);   // exact: < MTILES
    const int lane = threadIdx.x & 31;
    const int m    = lane & 15;
    const int koff = (lane >> 4) << 1;                        // 0 or 2
    const int row0 = wave * 16;
    const float* __restrict__ xrow = x + (size_t)(row0 + m) * NFEAT;

    v8f c0 = {}, c1 = {}, c2 = {}, c3 = {};
#pragma unroll 4
    for (int k0 = 0; k0 < NFEAT; k0 += 4) {
        v2f a;
        a.x = xrow[k0 + koff];
        a.y = xrow[k0 + koff + 1];
        const float* __restrict__ w0 = W1 + (size_t)(k0 + koff) * NHID;
        const float* __restrict__ w1 = W1 + (size_t)(k0 + koff + 1) * NHID;
        v2f b;
        b.x = w0[m];        b.y = w1[m];
        c0 = wmma_f32_4(a, b, c0);
        b.x = w0[m + 16];   b.y = w1[m + 16];
        c1 = wmma_f32_4(a, b, c1);
        b.x = w0[m + 32];   b.y = w1[m + 32];
        c2 = wmma_f32_4(a, b, c2);
        b.x = w0[m + 48];   b.y = w1[m + 48];
        c3 = wmma_f32_4(a, b, c3);
    }
    const int mb = (lane >> 4) * 8;            // C layout: vgpr r -> M=r (+8 hi lanes)
#pragma unroll
    for (int r = 0; r < 8; ++r) {
        float* __restrict__ drow = h + (size_t)(row0 + mb + r) * NHID + m;
        drow[0]  = c0[r];
        drow[16] = c1[r];
        drow[32] = c2[r];
        drow[48] = c3[r];
    }
}

// ---- self-loop term: agg = h * dinv^2 ---------------------------------------
__global__ void gcn_agg_init(const float* __restrict__ h, const float* __restrict__ dinv,
                             float* __restrict__ agg) {
    unsigned t = blockIdx.x * blockDim.x + threadIdx.x;
    if (t < (unsigned)GCN_N * NHID) {
        unsigned n = t >> 6;
        float di = dinv[n];
        agg[t] = h[t] * di * di;
    }
}

// ---- edge scatter: agg[dst] += h[src] * dinv[src]*dinv[dst] -----------------
__global__ void gcn_scatter(const float* __restrict__ h, const int* __restrict__ src,
                            const int* __restrict__ dst, const float* __restrict__ dinv,
                            float* __restrict__ agg) {
    unsigned t = blockIdx.x * blockDim.x + threadIdx.x;
    if (t >= (unsigned)GCN_E * NHID) return;
    unsigned e = t >> 6;
    int j = (int)(t & 63);
    int s = src[e];
    int d = dst[e];
    float nrm = dinv[s] * dinv[d];
    atomicAdd(&agg[(size_t)d * NHID + j], h[(size_t)s * NHID + j] * nrm);
}

// ---- epilogue: emb = relu(agg+b1); out = emb@Wc + bc; log_softmax -----------
__global__ void __launch_bounds__(TPB)
gcn_epilogue(const float* __restrict__ agg, const float* __restrict__ b1,
             const float* __restrict__ Wc, const float* __restrict__ bc,
             float* __restrict__ emb, float* __restrict__ lsm) {
    __shared__ float stage[WPB][16][48];
    const int wib  = threadIdx.x >> 5;
    const int lane = threadIdx.x & 31;
    const int tile = blockIdx.x * WPB + wib;                  // exact: < MTILES
    const int row0 = tile * 16;

    // store embedding = relu(agg + b1)
    for (int idx = lane; idx < 16 * NHID; idx += 32) {
        int mm = idx >> 6, kk = idx & 63;
        float v = agg[(size_t)(row0 + mm) * NHID + kk] + b1[kk];
        emb[(size_t)(row0 + mm) * NHID + kk] = fmaxf(v, 0.0f);
    }

    // GEMM2 tile: 16 x 48 (cols 40..47 are zero padding)
    const int m    = lane & 15;
    const int koff = (lane >> 4) << 1;
    const float* __restrict__ arow = agg + (size_t)(row0 + m) * NHID;
    v8f c0 = {}, c1 = {}, c2 = {};
    const int n2      = 32 + m;
    const bool n2ok   = (n2 < NCLASS);
#pragma unroll 4
    for (int k0 = 0; k0 < NHID; k0 += 4) {
        const int ka = k0 + koff, kb = ka + 1;
        v2f a;
        a.x = fmaxf(arow[ka] + b1[ka], 0.0f);
        a.y = fmaxf(arow[kb] + b1[kb], 0.0f);
        v2f b;
        b.x = Wc[(size_t)ka * NCLASS + m];
        b.y = Wc[(size_t)kb * NCLASS + m];
        c0 = wmma_f32_4(a, b, c0);
        b.x = Wc[(size_t)ka * NCLASS + 16 + m];
        b.y = Wc[(size_t)kb * NCLASS + 16 + m];
        c1 = wmma_f32_4(a, b, c1);
        b.x = n2ok ? Wc[(size_t)ka * NCLASS + n2] : 0.0f;
        b.y = n2ok ? Wc[(size_t)kb * NCLASS + n2] : 0.0f;
        c2 = wmma_f32_4(a, b, c2);
    }
    const int mb  = (lane >> 4) * 8;
    const float bc2 = n2ok ? bc[n2] : 0.0f;
#pragma unroll
    for (int r = 0; r < 8; ++r) {
        stage[wib][mb + r][m]      = c0[r] + bc[m];
        stage[wib][mb + r][16 + m] = c1[r] + bc[16 + m];
        stage[wib][mb + r][n2]     = c2[r] + bc2;
    }
    __syncthreads();

    // log-softmax: one thread per row (80 rows per block)
    if (threadIdx.x < WPB * 16) {
        const int w  = threadIdx.x >> 4;
        const int mm = threadIdx.x & 15;
        const int node = (blockIdx.x * WPB + w) * 16 + mm;
        const float* __restrict__ row = stage[w][mm];
        float mx = row[0];
        for (int j = 1; j < NCLASS; ++j) mx = fmaxf(mx, row[j]);
        float s = 0.0f;
        for (int j = 0; j < NCLASS; ++j) s += expf(row[j] - mx);
        const float lse = mx + logf(s);
        float* __restrict__ orow = lsm + (size_t)node * NCLASS;
        for (int j = 0; j < NCLASS; ++j) orow[j] = row[j] - lse;
    }
}

extern "C" void kernel_launch(void* const* d_in, const int* in_sizes, int n_in,
                              void* d_out, int out_size, void* d_ws, size_t ws_size,
                              hipStream_t stream) {
    (void)in_sizes; (void)n_in; (void)out_size; (void)ws_size;
    const float* x    = (const float*)d_in[0];
    const int*   edge = (const int*)d_in[1];          // [2,E] flat: src then dst
    const float* W1   = (const float*)d_in[2];
    const float* b1   = (const float*)d_in[3];
    const float* Wc   = (const float*)d_in[4];
    const float* bc   = (const float*)d_in[5];
    const int* src = edge;
    const int* dst = edge + GCN_E;

    float* out = (float*)d_out;
    float* emb = out;                                  // [N,64]
    float* lsm = out + (size_t)GCN_N * NHID;           // [N,40]
    float* h   = emb;                                  // stage h in d_out part1

    float* wsf  = (float*)d_ws;
    float* dinv = wsf;                                 // N floats
    float* agg  = wsf + 102400;                        // [N,64] fp32 (~25.6 MB)

    // 1-3: degree + symmetric normalization
    gcn_deg_init <<<(GCN_N + 255) / 256, 256, 0, stream>>>(dinv);
    gcn_deg_edges<<<(GCN_E + 255) / 256, 256, 0, stream>>>(dst, dinv);
    gcn_dinv     <<<(GCN_N + 255) / 256, 256, 0, stream>>>(dinv);

    // 4: h = x @ W1 (WMMA fp32)
    gcn_gemm1<<<MTILES / WPB, TPB, 0, stream>>>(x, W1, h);

    // 5: self-loop contribution
    gcn_agg_init<<<((unsigned)GCN_N * NHID + 255) / 256, 256, 0, stream>>>(h, dinv, agg);

    // 6: edge scatter (L2-resident gather + float atomics)
    gcn_scatter<<<((unsigned)GCN_E * NHID + 255) / 256, 256, 0, stream>>>(h, src, dst, dinv, agg);

    // 7: relu + classifier GEMM (WMMA fp32) + log_softmax
    gcn_epilogue<<<MTILES / WPB, TPB, 0, stream>>>(agg, b1, Wc, bc, emb, lsm);
}